// WaveMaskEncoder_4045859193025
// MI455X (gfx1250) — compile-verified
//
#include <hip/hip_runtime.h>

// ---------------------------------------------------------------------------
// Batched GEMM on the CDNA5 f32 WMMA path:
//   C[o, m, p] (+)= alpha * sum_k A[o, m, k] * W[k, p]
// Shapes: encode K=128,P=48  decode K=48,P=128 (template params).
// W is pre-interleaved once per launch into [k/2][p][2] order so each WMMA
// B-fragment is ONE aligned ds_load_b64; the interleaved W is staged into LDS
// with one Tensor-Data-Mover tensor_load_to_lds per block. Each wave computes
// a (16*MT) x (16*PT) macro-tile (MT=1|2, PT=3|4) -> 4..8 WMMAs per K-step.
// ---------------------------------------------------------------------------

typedef float v2f __attribute__((ext_vector_type(2)));
typedef float v8f __attribute__((ext_vector_type(8)));
typedef unsigned int u32x4 __attribute__((ext_vector_type(4)));
typedef int          i32x8 __attribute__((ext_vector_type(8)));
typedef int          i32x4 __attribute__((ext_vector_type(4)));

#define GAS __attribute__((address_space(1)))
typedef const GAS float* gcf_p;    // global const float*
typedef GAS float*       gf_p;     // global float*
typedef const GAS v2f*   gcv2f_p;  // global const v2f*

#define WAVES_PER_BLOCK 8
#define BLOCK_THREADS   256

// --- prep: Wi[(k/2)*(2P) + 2p + (k&1)] = W[k*P + p]  (pairs K for b64 B-frags)
__global__ __launch_bounds__(BLOCK_THREADS)
void interleave_w(const float* __restrict__ src, float* __restrict__ dst,
                  int K, int P)
{
    const int i = blockIdx.x * BLOCK_THREADS + threadIdx.x;
    if (i < K * P) {
        const int k = i / P;
        const int p = i - k * P;
        ((gf_p)dst)[(k >> 1) * (2 * P) + (p << 1) + (k & 1)] = ((gcf_p)src)[i];
    }
}

template <int KK, int PP, bool KCONTIG, int MT>
__global__ __launch_bounds__(BLOCK_THREADS)
void wmma_gemm_f32(const float* __restrict__ A,
                   const float* __restrict__ Wi,   // interleaved, KK*PP floats
                   float* __restrict__ C,
                   int M,
                   long long sAM, long long sAK,
                   long long sCM, long long sCP,
                   long long bsA, long long bsC,
                   int accFlag, float alpha)
{
    constexpr int PT  = (PP == 48) ? 3 : 4;  // 16-wide P tiles per wave
    constexpr int PW  = 16 * PT;             // P coverage per wave
    constexpr int TOT = KK * PP;             // 6144 floats = 24 KB

    __shared__ float sW[TOT];

    const int tid  = threadIdx.x;
    const int lane = tid & 31;
    const int wave = tid >> 5;

    const long long batch = blockIdx.y;
    const gcf_p Ab = (gcf_p)A + batch * bsA;
    const gf_p  Cb = (gf_p)C  + batch * bsC;

    // ---- Stage interleaved W into LDS via the Tensor Data Mover (flat 1-D
    // copy: tensor_dim0 = tile_dim0 = TOT, dim1 = 1).  D# per ISA ch.8.
    if (wave == 0) {
        const unsigned int lds = (unsigned int)(unsigned long long)(&sW[0]);
        const unsigned long long ga = (unsigned long long)Wi;
        u32x4 g0;
        g0.x = 1u;                                        // count=1, user desc
        g0.y = lds;                                       // lds_addr
        g0.z = (unsigned int)(ga & 0xFFFFFFFFull);        // global_addr[31:0]
        g0.w = (unsigned int)((ga >> 32) & 0x01FFFFFFull) // global_addr[56:32]
               | (2u << 30);                              // type = 2 ("image")
        i32x8 g1;
        g1[0] = 0x00020000;                   // data_size = 2 (4 bytes)
        g1[1] = (int)((TOT & 0xFFFF) << 16);  // tensor_dim0[15:0]
        g1[2] = (int)((TOT >> 16) | (1 << 16)); // dim0[31:16], tensor_dim1=1
        g1[3] = (int)(TOT << 16);             // dim1[31:16]=0, tile_dim0=TOT
        g1[4] = 1;                            // tile_dim1 = 1, tile_dim2 = 0
        g1[5] = TOT;                          // tensor_dim0_stride
        g1[6] = 0;
        g1[7] = 0;
        const i32x4 gz4 = {0, 0, 0, 0};
        const i32x8 gz8 = {0, 0, 0, 0, 0, 0, 0, 0};
        __builtin_amdgcn_tensor_load_to_lds(g0, g1, gz4, gz4, gz8, 0);
        __builtin_amdgcn_s_wait_tensorcnt(0);
    }
    __syncthreads();

    const int tilesPW = PP / PW;                 // 1 (encode) or 2 (decode)
    const int tilesMW = M / (16 * MT);           // launcher guarantees exact
    const int tile    = blockIdx.x * WAVES_PER_BLOCK + wave;
    if (tile >= tilesMW * tilesPW) return;       // wave-uniform exit

    const int tm = tile / tilesPW;
    const int tp = tile - tm * tilesPW;
    const int m0 = tm * (16 * MT);
    const int p0 = tp * PW;

    // Fragment lane mapping (ISA 7.12.2, 32-bit 16x4 A / 4x16 B / 16x16 C):
    const int laneM = lane & 15;
    const int kOff  = (lane >> 4) << 1;          // lanes 16-31 hold K+2,K+3

    gcf_p ap[MT];
#pragma unroll
    for (int mt = 0; mt < MT; ++mt)
        ap[mt] = Ab + (long long)(m0 + mt * 16 + laneM) * sAM
                    + (long long)kOff * sAK;
    const long long aStep = 4LL * sAK;
    // Interleaved B: pair-row r = (k0+kOff)/2, elem = sW[r*2*PP + 2*col + 0/1]
    int bIdx = kOff * PP + 2 * (p0 + laneM);

    v8f acc[MT][PT];
#pragma unroll
    for (int mt = 0; mt < MT; ++mt)
#pragma unroll
        for (int pt = 0; pt < PT; ++pt)
            acc[mt][pt] = v8f{0.f, 0.f, 0.f, 0.f, 0.f, 0.f, 0.f, 0.f};

#pragma unroll 4
    for (int k0 = 0; k0 < KK; k0 += 4) {
        v2f a[MT];
        v2f b[PT];
#pragma unroll
        for (int mt = 0; mt < MT; ++mt) {
            if (KCONTIG) {
                a[mt] = *(gcv2f_p)ap[mt];     // one global_load_b64 (sAK==1)
            } else {
                a[mt].x = ap[mt][0];
                a[mt].y = ap[mt][sAK];
            }
        }
#pragma unroll
        for (int pt = 0; pt < PT; ++pt)
            b[pt] = *(const v2f*)&sW[bIdx + pt * 32];   // one ds_load_b64
#pragma unroll
        for (int mt = 0; mt < MT; ++mt)
#pragma unroll
            for (int pt = 0; pt < PT; ++pt)
                acc[mt][pt] = __builtin_amdgcn_wmma_f32_16x16x4_f32(
                    false, a[mt], false, b[pt], (short)0,
                    acc[mt][pt], false, false);
#pragma unroll
        for (int mt = 0; mt < MT; ++mt) ap[mt] += aStep;
        bIdx += 4 * PP;
    }

    // C/D layout: lanes 0-15 -> N=lane, rows v; lanes 16-31 -> rows v+8.
#pragma unroll
    for (int mt = 0; mt < MT; ++mt) {
        const int rbase = m0 + mt * 16 + ((lane >> 4) << 3);
#pragma unroll
        for (int pt = 0; pt < PT; ++pt) {
            const int cc = p0 + pt * 16 + laneM;
#pragma unroll
            for (int v = 0; v < 8; ++v) {
                const long long idx =
                    (long long)(rbase + v) * sCM + (long long)cc * sCP;
                const float val = alpha * acc[mt][pt][v];
                if (accFlag) Cb[idx] += val;
                else         Cb[idx]  = val;
            }
        }
    }
}

// ---------------------------------------------------------------------------

static inline void launch_gemm(const float* A, const float* Wi, float* C,
                               int M, int K, int P,
                               long long sAM, long long sAK,
                               long long sCM, long long sCP,
                               long long bsA, long long bsC,
                               int batches, int accFlag, float alpha,
                               hipStream_t stream)
{
    const bool encode  = (K == 128);             // encode: K=128,P=48
    const bool m32     = (M % 32 == 0);          // false only for D2a (M=48)
    const int  PW      = encode ? 48 : 64;
    const int  tilesMW = M / (m32 ? 32 : 16);
    const int  tiles   = tilesMW * (P / PW);
    dim3 grid((tiles + WAVES_PER_BLOCK - 1) / WAVES_PER_BLOCK, batches);
    const bool kcontig = (sAK == 1);

    if (encode) {
        if (kcontig)
            wmma_gemm_f32<128, 48, true, 2><<<grid, BLOCK_THREADS, 0, stream>>>(
                A, Wi, C, M, sAM, sAK, sCM, sCP, bsA, bsC, accFlag, alpha);
        else
            wmma_gemm_f32<128, 48, false, 2><<<grid, BLOCK_THREADS, 0, stream>>>(
                A, Wi, C, M, sAM, sAK, sCM, sCP, bsA, bsC, accFlag, alpha);
    } else if (kcontig) {
        wmma_gemm_f32<48, 128, true, 2><<<grid, BLOCK_THREADS, 0, stream>>>(
            A, Wi, C, M, sAM, sAK, sCM, sCP, bsA, bsC, accFlag, alpha);
    } else if (m32) {
        wmma_gemm_f32<48, 128, false, 2><<<grid, BLOCK_THREADS, 0, stream>>>(
            A, Wi, C, M, sAM, sAK, sCM, sCP, bsA, bsC, accFlag, alpha);
    } else {
        wmma_gemm_f32<48, 128, false, 1><<<grid, BLOCK_THREADS, 0, stream>>>(
            A, Wi, C, M, sAM, sAK, sCM, sCP, bsA, bsC, accFlag, alpha);
    }
}

extern "C" void kernel_launch(void* const* d_in, const int* in_sizes, int n_in,
                              void* d_out, int out_size, void* d_ws, size_t ws_size,
                              hipStream_t stream)
{
    (void)in_sizes; (void)n_in; (void)out_size; (void)ws_size;

    // B=2, C=8, L=128, N=48
    const float* x   = (const float*)d_in[0];
    const float* EN3 = (const float*)d_in[1];
    const float* EN2 = (const float*)d_in[2];
    const float* EN1 = (const float*)d_in[3];
    const float* DE3 = (const float*)d_in[4];
    const float* DE2 = (const float*)d_in[5];
    const float* DE1 = (const float*)d_in[6];
    float* out = (float*)d_out;

    // Workspace (floats): e1 | e2 | enc | 6 interleaved W matrices
    float* e1  = (float*)d_ws;             // 16*48*128*128 = 12,582,912
    float* e2b = e1  + 12582912LL;         // 16*48*48*128  =  4,718,592
    float* enc = e2b + 4718592LL;          // 16*48*48*48   =  1,769,472
    float* wi  = enc + 1769472LL;          // 6 * 6144
    float* wEN3 = wi;
    float* wEN2 = wi + 6144;
    float* wEN1 = wi + 12288;
    float* wDE3 = wi + 18432;
    float* wDE2 = wi + 24576;
    float* wDE1 = wi + 30720;

    // ---- pre-interleave the six small matrices (K-pairs adjacent) ----
    {
        dim3 g(24);  // 6144 / 256
        interleave_w<<<g, BLOCK_THREADS, 0, stream>>>(EN3, wEN3, 128, 48);
        interleave_w<<<g, BLOCK_THREADS, 0, stream>>>(EN2, wEN2, 128, 48);
        interleave_w<<<g, BLOCK_THREADS, 0, stream>>>(EN1, wEN1, 128, 48);
        interleave_w<<<g, BLOCK_THREADS, 0, stream>>>(DE3, wDE3, 48, 128);
        interleave_w<<<g, BLOCK_THREADS, 0, stream>>>(DE2, wDE2, 48, 128);
        interleave_w<<<g, BLOCK_THREADS, 0, stream>>>(DE1, wDE1, 48, 128);
    }

    const float third = 1.0f / 3.0f;

    // ---------------- encode ----------------
    // E1: bcdhw,dp->bcphw   (contract d, Inner = L*L)
    launch_gemm(x,   wEN3, e1, 16384, 128, 48, 1, 16384, 1, 16384,
                2097152LL, 786432LL, 16, 0, 1.0f, stream);
    // E2: bcphw,hq->bcpqw   (contract h, Inner = L)
    launch_gemm(e1,  wEN2, e2b, 128, 128, 48, 1, 128, 1, 128,
                16384LL, 6144LL, 768, 0, 1.0f, stream);
    // E3: bcpqw,wr->bcpqr   (contract last axis w)
    launch_gemm(e2b, wEN1, enc, 36864, 128, 48, 128, 1, 48, 1,
                0LL, 0LL, 1, 0, 1.0f, stream);

    // ---------------- decode, order W,H,D ----------------
    // D1a: bcpqr,rw->bcpqw
    launch_gemm(enc, wDE1, e2b, 36864, 48, 128, 48, 1, 128, 1,
                0LL, 0LL, 1, 0, 1.0f, stream);
    // D1b: bcpqw,qh->bcphw  (contract q, Inner = L)
    launch_gemm(e2b, wDE2, e1, 128, 48, 128, 1, 128, 1, 128,
                6144LL, 16384LL, 768, 0, 1.0f, stream);
    // D1c: bcphw,pd->bcdhw  (contract p, Inner = L*L) -> out = val/3
    launch_gemm(e1,  wDE3, out, 16384, 48, 128, 1, 16384, 1, 16384,
                786432LL, 2097152LL, 16, 0, third, stream);

    // ---------------- decode, order H,D,W ----------------
    // D2a: bcpqr,qh->bcphr  (contract q, Inner = N)  [M=48 -> MT=1 variant]
    launch_gemm(enc, wDE2, e2b, 48, 48, 128, 1, 48, 1, 48,
                2304LL, 6144LL, 768, 0, 1.0f, stream);
    // D2b: bcphr,pd->bcdhr  (contract p, Inner = L*N)
    launch_gemm(e2b, wDE3, e1, 6144, 48, 128, 1, 6144, 1, 6144,
                294912LL, 786432LL, 16, 0, 1.0f, stream);
    // D2c: bcdhr,rw->bcdhw  (contract last axis r) -> out += val/3
    launch_gemm(e1,  wDE1, out, 262144, 48, 128, 48, 1, 128, 1,
                0LL, 0LL, 1, 1, third, stream);

    // ---------------- decode, order D,W,H ----------------
    // D3a: bcpqr,pd->bcdqr  (contract p, Inner = N*N)
    launch_gemm(enc, wDE3, e2b, 2304, 48, 128, 1, 2304, 1, 2304,
                110592LL, 294912LL, 16, 0, 1.0f, stream);
    // D3b: bcdqr,rw->bcdqw  (contract last axis r)
    launch_gemm(e2b, wDE1, e1, 98304, 48, 128, 48, 1, 128, 1,
                0LL, 0LL, 1, 0, 1.0f, stream);
    // D3c: bcdqw,qh->bcdhw  (contract q, Inner = L) -> out += val/3
    launch_gemm(e1,  wDE2, out, 128, 48, 128, 1, 128, 1, 128,
                6144LL, 16384LL, 2048, 1, third, stream);
}